// UV_Rasterizer_15668040695949
// MI455X (gfx1250) — compile-verified
//
#include <hip/hip_runtime.h>
#include <hip/hip_bf16.h>
#include <math.h>

typedef __attribute__((ext_vector_type(2))) float v2f;
typedef __attribute__((ext_vector_type(8))) float v8f;

#define UV 256
#define EPSR 1e-8f

// ---------------------------------------------------------------------------
// Pass 1: per-face affine coefficients, written in WMMA A-matrix order.
// Chunk c = faces 4c..4c+3. Within a chunk (64 floats):
//   float index (m*2 + k)        : m in 0..15 (= faceInChunk*4 + {w0,w1,w2,z}),
//                                  k=0 -> x-coeff, k=1 -> y-coeff   (lanes 0..15)
//   float index 32 + (m*2 + k)   : k=0 -> const coeff, k=1 -> 0     (lanes 16..31)
// Raster wave loads its A operand with one b64:  v2f A = coef2[c*32 + lane];
// ---------------------------------------------------------------------------
__global__ void uvr_precompute(const float* __restrict__ verts,
                               const long long* __restrict__ faces,
                               float* __restrict__ coef, int F) {
    int g = blockIdx.x * blockDim.x + threadIdx.x;
    if (g >= F) return;

    int i0 = (int)faces[3 * g + 0];
    int i1 = (int)faces[3 * g + 1];
    int i2 = (int)faces[3 * g + 2];

    float x0 = verts[3 * i0 + 0], y0 = verts[3 * i0 + 1], z0 = verts[3 * i0 + 2];
    float x1 = verts[3 * i1 + 0], y1 = verts[3 * i1 + 1], z1 = verts[3 * i1 + 2];
    float x2 = verts[3 * i2 + 0], y2 = verts[3 * i2 + 1], z2 = verts[3 * i2 + 2];

    float den = (y1 - y2) * (x0 - x2) + (x2 - x1) * (y0 - y2);
    bool ok = fabsf(den) > EPSR;
    float ds = ok ? den : EPSR;

    float a0 = (y1 - y2) / ds, b0 = (x2 - x1) / ds;
    float c0 = -(a0 * x2 + b0 * y2);
    float a1 = (y2 - y0) / ds, b1 = (x0 - x2) / ds;
    float c1 = -(a1 * x2 + b1 * y2);
    float a2 = -a0 - a1, b2 = -b0 - b1, c2 = 1.0f - c0 - c1;

    float az = a0 * z0 + a1 * z1 + a2 * z2;
    float bz = b0 * z0 + b1 * z1 + b2 * z2;
    float cz = c0 * z0 + c1 * z1 + c2 * z2;

    if (!ok) {  // degenerate: force w0 == -1 everywhere -> never inside
        a0 = b0 = 0.0f; c0 = -1.0f;
        a1 = b1 = c1 = 0.0f;
        a2 = b2 = c2 = 0.0f;
        az = bz = cz = 0.0f;
    }

    int c  = g >> 2;
    int fi = g & 3;
    float* cp = coef + (size_t)c * 64;
    int m = fi * 4;

    // x/y coefficient pairs (A-matrix lanes 0..15, K=0,1)
    cp[(m + 0) * 2 + 0] = a0;  cp[(m + 0) * 2 + 1] = b0;
    cp[(m + 1) * 2 + 0] = a1;  cp[(m + 1) * 2 + 1] = b1;
    cp[(m + 2) * 2 + 0] = a2;  cp[(m + 2) * 2 + 1] = b2;
    cp[(m + 3) * 2 + 0] = az;  cp[(m + 3) * 2 + 1] = bz;
    // constant terms (A-matrix lanes 16..31, K=2,3)
    cp[32 + (m + 0) * 2 + 0] = c0;  cp[32 + (m + 0) * 2 + 1] = 0.0f;
    cp[32 + (m + 1) * 2 + 0] = c1;  cp[32 + (m + 1) * 2 + 1] = 0.0f;
    cp[32 + (m + 2) * 2 + 0] = c2;  cp[32 + (m + 2) * 2 + 1] = 0.0f;
    cp[32 + (m + 3) * 2 + 0] = cz;  cp[32 + (m + 3) * 2 + 1] = 0.0f;
}

// ---------------------------------------------------------------------------
// Pass 2: each wave32 owns 16 pixels. One V_WMMA_F32_16X16X4_F32 per 4-face
// chunk evaluates (w0,w1,w2,z) of 4 faces at all 16 pixels. D-layout puts
// faces {4c,4c+1} in lanes 0..15 and {4c+2,4c+3} in lanes 16..31, so the
// z-buffer update is in-lane; a final shfl_xor(16) merges the halves.
// NCHUNKS > 0 -> compile-time trip count (no remainder loop).
// ---------------------------------------------------------------------------
template <int NCHUNKS>
__global__ __launch_bounds__(256) void uvr_raster(
    const float* __restrict__ coef, const float* __restrict__ attrs,
    float* __restrict__ out, int nchunks_dyn) {
    const int lane = threadIdx.x & 31;
    const int wave = blockIdx.x * (blockDim.x >> 5) + (threadIdx.x >> 5);
    const int l15  = lane & 15;
    const int pix  = wave * 16 + l15;          // 16 consecutive pixels per wave
    const int ii   = pix >> 8;                 // row
    const int jj   = pix & 255;                // col

    const float px = 2.0f * ((float)jj + 0.5f) * (1.0f / (float)UV) - 1.0f;
    const float py = 2.0f * ((float)ii + 0.5f) * (1.0f / (float)UV) - 1.0f;

    // B matrix (4 x 16): rows PX, PY, 1, 0. K=0,2 in B[0]; K=1,3 in B[1].
    v2f B;
    B[0] = (lane < 16) ? px : 1.0f;
    B[1] = (lane < 16) ? py : 0.0f;

    float zb = INFINITY;
    int   bi = -1;
    const int fofs = (lane >= 16) ? 2 : 0;
    const v2f* __restrict__ cp2 = (const v2f*)coef;
    const int nchunks = (NCHUNKS > 0) ? NCHUNKS : nchunks_dyn;

#pragma unroll 4
    for (int c = 0; c < nchunks; ++c) {
        v2f A = cp2[(size_t)c * 32 + lane];
        __builtin_prefetch(&cp2[(size_t)(c + 16) * 32 + lane], 0, 3);
        v8f acc = {};
        v8f d = __builtin_amdgcn_wmma_f32_16x16x4_f32(
            false, A, false, B, (short)0, acc, false, false);
        int f0 = 4 * c + fofs;
        // inside test as a min-reduction -> v_min3/v_min_num instead of
        // three v_cmp + scalar ANDs.
        float m0 = fminf(fminf(d[0], d[1]), d[2]);
        float m1 = fminf(fminf(d[4], d[5]), d[6]);
        if ((m0 >= 0.0f) && (d[3] < zb)) { zb = d[3]; bi = f0; }
        if ((m1 >= 0.0f) && (d[7] < zb)) { zb = d[7]; bi = f0 + 1; }
    }

    // Merge the two half-wave candidates for this pixel (tie -> smaller index).
    float zo = __shfl_xor(zb, 16, 32);
    int   io = __shfl_xor(bi, 16, 32);
    if ((zo < zb) || (zo == zb && (unsigned)io < (unsigned)bi)) { zb = zo; bi = io; }

    if (lane < 16) {
        float r = 0.0f, g = 0.0f, b = 0.0f, vis = 0.0f;
        if (bi >= 0) {
            vis = 1.0f;
            int cc = bi >> 2, fi = bi & 3;
            const float* cb = coef + (size_t)cc * 64 + fi * 8;       // (a,b) pairs
            const float* kb = coef + (size_t)cc * 64 + 32 + fi * 8;  // const terms
            float w0 = fmaf(cb[0], px, fmaf(cb[1], py, kb[0]));
            float w1 = fmaf(cb[2], px, fmaf(cb[3], py, kb[2]));
            float w2 = fmaf(cb[4], px, fmaf(cb[5], py, kb[4]));
            const float* ap = attrs + (size_t)bi * 9;                // [3 verts][3]
            r = w0 * ap[0] + w1 * ap[3] + w2 * ap[6];
            g = w0 * ap[1] + w1 * ap[4] + w2 * ap[7];
            b = w0 * ap[2] + w1 * ap[5] + w2 * ap[8];
        }
        out[0 * (UV * UV) + pix] = r;
        out[1 * (UV * UV) + pix] = g;
        out[2 * (UV * UV) + pix] = b;
        out[3 * (UV * UV) + pix] = vis;
    }
}

extern "C" void kernel_launch(void* const* d_in, const int* in_sizes, int n_in,
                              void* d_out, int out_size, void* d_ws, size_t ws_size,
                              hipStream_t stream) {
    const float*     verts = (const float*)d_in[0];      // [1,V,3] f32
    const long long* faces = (const long long*)d_in[1];  // [1,F,3] i64
    const float*     attrs = (const float*)d_in[2];      // [1,F,3,3] f32
    float* out  = (float*)d_out;                         // [1,4,256,256] f32
    float* coef = (float*)d_ws;                          // F*16 floats (512 KB)

    int F = in_sizes[2] / 9;                             // faces count
    int nchunks = F >> 2;                                // 4 faces per WMMA

    uvr_precompute<<<(F + 255) / 256, 256, 0, stream>>>(verts, faces, coef, F);

    int waves  = (UV * UV) / 16;                         // 4096 waves
    int blocks = waves / 8;                              // 8 waves / block
    if (nchunks == 2048) {
        uvr_raster<2048><<<blocks, 256, 0, stream>>>(coef, attrs, out, nchunks);
    } else {
        uvr_raster<0><<<blocks, 256, 0, stream>>>(coef, attrs, out, nchunks);
    }
}